// GraphOfGraphsLayer_7232724927022
// MI455X (gfx1250) — compile-verified
//
#include <hip/hip_runtime.h>
#include <hip/hip_bf16.h>

typedef __attribute__((ext_vector_type(2))) float v2f;
typedef __attribute__((ext_vector_type(8))) float v8f;

#define WMMA_F32X4(a_, b_, c_) \
  __builtin_amdgcn_wmma_f32_16x16x4_f32(false, (a_), false, (b_), (short)0, (c_), false, false)

#define GN 1024
#define GD 256
#define GH 128
#define HPAD 129   // 129 % 64 == 1 -> conflict-free strided LDS reads
#define KSPLIT 8   // split-K factor for the 1024x1024x128 GEMM (64 -> 512 waves)

// ---------------------------------------------------------------------------
// Kernel 1: strip GEMM  C[1024,128] = X[1024,256] @ W[256,128]  (+b1 for z==0)
// One wave per 16-row strip; 8 accumulator tiles cover all H=128 columns.
// blockIdx.y selects which projection: 0 -> a (=X@W1[:D]+b1), 1 -> c, 2 -> g.
// ---------------------------------------------------------------------------
__global__ void __launch_bounds__(32) k_gemm_xw(
    const float* __restrict__ X, const float* __restrict__ W1,
    const float* __restrict__ Wg, const float* __restrict__ b1,
    float* __restrict__ a_out, float* __restrict__ c_out, float* __restrict__ g_out)
{
  const int lane = threadIdx.x;
  const int m0 = blockIdx.x * 16;
  const int z = blockIdx.y;
  const float* W = (z == 0) ? W1 : (z == 1 ? (W1 + GD * GH) : Wg);
  float* Out = (z == 0) ? a_out : (z == 1 ? c_out : g_out);

  const int row   = lane & 15;
  const int khalf = (lane >> 4) << 1;   // 0 or 2
  const int col   = lane & 15;

  v8f acc[8] = {};
  for (int k0 = 0; k0 < GD; k0 += 4) {
    const int kk = k0 + khalf;
    const float* xp = X + (m0 + row) * GD + kk;
    v2f af; af.x = xp[0]; af.y = xp[1];
#pragma unroll
    for (int t = 0; t < 8; ++t) {
      const float* wp = W + kk * GH + t * 16 + col;
      v2f bf; bf.x = wp[0]; bf.y = wp[GH];
      acc[t] = WMMA_F32X4(af, bf, acc[t]);
    }
  }
  const int mbase = m0 + ((lane >> 4) << 3);
#pragma unroll
  for (int t = 0; t < 8; ++t) {
    const int n = t * 16 + col;
    const float bias = (z == 0) ? b1[n] : 0.0f;
#pragma unroll
    for (int r = 0; r < 8; ++r)
      Out[(mbase + r) * GH + n] = acc[t][r] + bias;
  }
}

// ---------------------------------------------------------------------------
// Kernel 2: pairwise similarity -> adjacency A (upper-tri tiles only, mirrored)
// The ReLU prevents matmul structure: this is the irreducible N^2*H VALU core.
// 16x16 (i,j) tile per 256-thread block, a/c tiles staged in padded LDS.
// ---------------------------------------------------------------------------
__global__ void __launch_bounds__(256) k_sim(
    const float* __restrict__ a, const float* __restrict__ c,
    const float* __restrict__ W2, const float* __restrict__ b2,
    float* __restrict__ A)
{
  const int ti = blockIdx.y, tj = blockIdx.x;
  if (tj < ti) return;

  __shared__ float sA[16 * HPAD];
  __shared__ float sC[16 * HPAD];
  __shared__ float sW[GH];
  const int tid = threadIdx.x;

  for (int idx = tid; idx < 16 * GH; idx += 256) {
    const int r = idx >> 7, h = idx & (GH - 1);
    sA[r * HPAD + h] = a[(ti * 16 + r) * GH + h];
    sC[r * HPAD + h] = c[(tj * 16 + r) * GH + h];
  }
  if (tid < GH) sW[tid] = W2[tid];
  __syncthreads();

  const int tx = tid & 15;   // local j
  const int ty = tid >> 4;   // local i
  const float* pa = &sA[ty * HPAD];
  const float* pc = &sC[tx * HPAD];
  float zz = 0.0f;
#pragma unroll 8
  for (int h = 0; h < GH; ++h)
    zz = fmaf(fmaxf(pa[h] + pc[h], 0.0f), sW[h], zz);
  zz += b2[0];

  const float s = 1.0f / (1.0f + __expf(-zz));
  const int gi = ti * 16 + ty, gj = tj * 16 + tx;
  if (gi < gj) {
    const float val = (s > 0.5f) ? s : 0.0f;
    A[gi * GN + gj] = val;
    A[gj * GN + gi] = val;
  } else if (gi == gj) {
    A[gi * GN + gi] = 1.0f;
  }
}

// ---------------------------------------------------------------------------
// Kernel 3: deterministic degree row-sums -> dis[i] = rsqrt(deg[i])
// (no float atomics -> bit-stable across graph replays)
// ---------------------------------------------------------------------------
__global__ void __launch_bounds__(256) k_deg(const float* __restrict__ A,
                                             float* __restrict__ dis)
{
  const int i = blockIdx.x, tid = threadIdx.x;
  const float* row = A + i * GN;
  float s = 0.0f;
  for (int j = tid; j < GN; j += 256) s += row[j];
  __shared__ float red[256];
  red[tid] = s; __syncthreads();
  for (int off = 128; off > 0; off >>= 1) {
    if (tid < off) red[tid] += red[tid + off];
    __syncthreads();
  }
  if (tid == 0) dis[i] = rsqrtf(red[0]);
}

// Kernel 4: gp[k,:] = dis[k] * g[k,:]
__global__ void __launch_bounds__(256) k_scale_g(const float* __restrict__ g,
                                                 const float* __restrict__ dis,
                                                 float* __restrict__ gp)
{
  const int idx = blockIdx.x * 256 + threadIdx.x;
  gp[idx] = g[idx] * dis[idx >> 7];
}

// ---------------------------------------------------------------------------
// Kernel 5: split-K WMMA GEMM partials:
//   part[kc][m,:] = A[m, kc*128 : kc*128+128] @ gp[kc*128 : kc*128+128, :]
// Grid (64 M-tiles, 8 K-chunks) = 512 waves -> fills MI455X's WGPs instead of
// 64 serial single-wave chains. 256 v_wmma_f32_16x16x4_f32 per wave.
// ---------------------------------------------------------------------------
__global__ void __launch_bounds__(32) k_gemm_ag_part(
    const float* __restrict__ A, const float* __restrict__ gp,
    float* __restrict__ part)
{
  const int lane = threadIdx.x;
  const int m0 = blockIdx.x * 16;
  const int kc = blockIdx.y;
  const int kbeg = kc * (GN / KSPLIT);
  const int kend = kbeg + (GN / KSPLIT);

  const int row   = lane & 15;
  const int khalf = (lane >> 4) << 1;
  const int col   = lane & 15;

  v8f acc[8] = {};
  for (int k0 = kbeg; k0 < kend; k0 += 4) {
    const int kk = k0 + khalf;
    const float* ap = A + (m0 + row) * GN + kk;
    v2f af; af.x = ap[0]; af.y = ap[1];
#pragma unroll
    for (int t = 0; t < 8; ++t) {
      const float* bp = gp + kk * GH + t * 16 + col;
      v2f bf; bf.x = bp[0]; bf.y = bp[GH];
      acc[t] = WMMA_F32X4(af, bf, acc[t]);
    }
  }
  float* pout = part + kc * (GN * GH);
  const int mbase = m0 + ((lane >> 4) << 3);
#pragma unroll
  for (int t = 0; t < 8; ++t) {
    const int n = t * 16 + col;
#pragma unroll
    for (int r = 0; r < 8; ++r)
      pout[(mbase + r) * GH + n] = acc[t][r];
  }
}

// Kernel 5b: fixed-order reduce of the 8 partials + dis_i scale + bg bias.
__global__ void __launch_bounds__(256) k_reduce_ag(
    const float* __restrict__ part, const float* __restrict__ dis,
    const float* __restrict__ bg, float* __restrict__ pre)
{
  const int idx = blockIdx.x * 256 + threadIdx.x;   // < N*H
  float s = 0.0f;
#pragma unroll
  for (int w = 0; w < KSPLIT; ++w) s += part[w * (GN * GH) + idx];
  const int m = idx >> 7;
  pre[idx] = fmaf(s, dis[m], bg[idx & (GH - 1)]);
}

// ---------------------------------------------------------------------------
// Kernel 6: BatchNorm over rows (biased var) + affine + ReLU, one block/column.
// ---------------------------------------------------------------------------
__global__ void __launch_bounds__(256) k_bn(
    const float* __restrict__ pre, const float* __restrict__ gamma,
    const float* __restrict__ beta, float* __restrict__ out)
{
  const int h = blockIdx.x, tid = threadIdx.x;
  float v[4], s = 0.0f, sq = 0.0f;
#pragma unroll
  for (int r = 0; r < 4; ++r) {
    v[r] = pre[(tid + r * 256) * GH + h];
    s += v[r]; sq += v[r] * v[r];
  }
  __shared__ float rs[256], rq[256];
  rs[tid] = s; rq[tid] = sq; __syncthreads();
  for (int off = 128; off > 0; off >>= 1) {
    if (tid < off) { rs[tid] += rs[tid + off]; rq[tid] += rq[tid + off]; }
    __syncthreads();
  }
  __shared__ float sscale, sshift;
  if (tid == 0) {
    const float mean = rs[0] * (1.0f / 1024.0f);
    const float var  = rq[0] * (1.0f / 1024.0f) - mean * mean;
    const float sc   = gamma[h] * rsqrtf(var + 1e-5f);
    sscale = sc; sshift = beta[h] - mean * sc;
  }
  __syncthreads();
#pragma unroll
  for (int r = 0; r < 4; ++r)
    out[(tid + r * 256) * GH + h] = fmaxf(fmaf(v[r], sscale, sshift), 0.0f);
}

extern "C" void kernel_launch(void* const* d_in, const int* in_sizes, int n_in,
                              void* d_out, int out_size, void* d_ws, size_t ws_size,
                              hipStream_t stream)
{
  (void)in_sizes; (void)n_in; (void)out_size; (void)ws_size;
  const float* X     = (const float*)d_in[0];  // [1024,256]
  const float* W1    = (const float*)d_in[1];  // [512,128]
  const float* b1    = (const float*)d_in[2];  // [128]
  const float* W2    = (const float*)d_in[3];  // [128,1]
  const float* b2    = (const float*)d_in[4];  // [1]
  const float* Wg    = (const float*)d_in[5];  // [256,128]
  const float* bg    = (const float*)d_in[6];  // [128]
  const float* gamma = (const float*)d_in[7];  // [128]
  const float* beta  = (const float*)d_in[8];  // [128]
  float* out = (float*)d_out;                  // [1024,128]

  // Workspace layout (floats): A | a | c | g | gp | dis | pre | partials (~11 MB)
  float* ws   = (float*)d_ws;
  float* Amat = ws;                          // 1024*1024
  float* av   = Amat + GN * GN;              // 1024*128
  float* cv   = av + GN * GH;
  float* gv   = cv + GN * GH;
  float* gpv  = gv + GN * GH;
  float* disv = gpv + GN * GH;               // 1024
  float* prev = disv + GN;                   // 1024*128
  float* partv = prev + GN * GH;             // KSPLIT*1024*128

  k_gemm_xw<<<dim3(GN / 16, 3), 32, 0, stream>>>(X, W1, Wg, b1, av, cv, gv);
  k_sim<<<dim3(GN / 16, GN / 16), 256, 0, stream>>>(av, cv, W2, b2, Amat);
  k_deg<<<GN, 256, 0, stream>>>(Amat, disv);
  k_scale_g<<<(GN * GH) / 256, 256, 0, stream>>>(gv, disv, gpv);
  k_gemm_ag_part<<<dim3(GN / 16, KSPLIT), 32, 0, stream>>>(Amat, gpv, partv);
  k_reduce_ag<<<(GN * GH) / 256, 256, 0, stream>>>(partv, disv, bg, prev);
  k_bn<<<GH, 256, 0, stream>>>(prev, gamma, beta, out);
}